// VoxelSubdivision_91336774517360
// MI455X (gfx1250) — compile-verified
//
#include <hip/hip_runtime.h>

// VoxelSubdivision for MI455X (gfx1250, wave32).
//
// Slab test factored as rank-2 outer products -> V_WMMA_F32_16X16X4_F32:
//   t1[r][v] = inv_a[r]*c_a[v] + beta1_a[r],  beta1 = -(o+half)*inv
//   t2[r][v] = inv_a[r]*c_a[v] + beta2_a[r],  beta2 = (half-o)*inv
// A row (per ray)  = [inv, beta, 0, 0]   (16x4 f32 -> v2f per lane)
// B col (per voxel)= [c,   1,    0, 0]   (4x16 f32 -> v2f per lane)
// D (16 rays x 16 voxels f32) -> v8f per lane; element i is row (lane<16?i:8+i),
// column lane&15.
//
// A line crosses <= 3*16-2 = 46 voxels of a 16^3 grid, so per-ray hits are
// compacted into a 64-entry LDS list (ds_add_rtn_u32 slot allocation) as
// 64-bit keys (orderedFloat(depth)<<32 | voxelIdx). Top-31 = 31 rounds of a
// wave32 shfl_xor u64-min butterfly over 2 registers/lane; pads decode to
// (MISS, -1), reproducing jax.lax.top_k ordering + the reference's masking.

#define NVOX        4096
#define RAYS_PER_WG 16
#define KHITS       31
#define CAP         64          // >= 46 = max voxels a line can cross
#define MISS_F      10000.0f
#define VOXSZ       (2.0f / 15.0f + 1e-08f)

typedef __attribute__((ext_vector_type(2))) float v2f;
typedef __attribute__((ext_vector_type(8))) float v8f;

static __device__ __forceinline__ v8f wmma4(v2f a, v2f b) {
  v8f c = {};
  // (neg_a, A, neg_b, B, c_mod, C, reuse_a, reuse_b)
  return __builtin_amdgcn_wmma_f32_16x16x4_f32(false, a, false, b, (short)0, c,
                                               false, false);
}

// Monotone map: float compare == unsigned compare.
static __device__ __forceinline__ unsigned long long packKey(float d, int idx) {
  unsigned int b = __float_as_uint(d);
  unsigned int u = b ^ ((b & 0x80000000u) ? 0xFFFFFFFFu : 0x80000000u);
  return ((unsigned long long)u << 32) | (unsigned int)idx;
}

__global__ __launch_bounds__(256) void voxel_subdiv_kernel(
    const float* __restrict__ rays,      // [R,6]
    const float* __restrict__ centers,   // [4096,3]
    float* __restrict__ out,             // concatenated outputs (see below)
    int numRays) {
  __shared__ unsigned long long sKey[RAYS_PER_WG][CAP];  // 8 KB hit lists
  __shared__ int   sCnt[RAYS_PER_WG];
  __shared__ float sResD[RAYS_PER_WG][KHITS];
  __shared__ int   sResI[RAYS_PER_WG][KHITS];

  const int tid   = threadIdx.x;
  const int wave  = tid >> 5;
  const int lane  = tid & 31;
  const int row   = lane & 15;
  const bool lo   = (lane < 16);
  const int rBase = blockIdx.x * RAYS_PER_WG;

  if (tid < RAYS_PER_WG) sCnt[tid] = 0;

  // Warm caches for the (tiny, fully reused) voxel-center table.
  __builtin_prefetch(centers + (tid & 63) * 192, 0, 1);

  // ---- per-ray WMMA A fragments (identical in every wave of the block) ----
  const float hf = VOXSZ * 0.5f;
  float o[3], dr[3], inv[3];
  {
    const float* rp = rays + (size_t)(rBase + row) * 6;
#pragma unroll
    for (int a = 0; a < 3; ++a) {
      o[a]   = rp[a];
      dr[a]  = rp[3 + a];
      inv[a] = 1.0f / dr[a];
    }
  }
  v2f A1[3], A2[3];
#pragma unroll
  for (int a = 0; a < 3; ++a) {
    float b1 = -(o[a] + hf) * inv[a];
    float b2 = (hf - o[a]) * inv[a];
    A1[a].x = lo ? inv[a] : 0.0f;  A1[a].y = lo ? b1 : 0.0f;
    A2[a].x = lo ? inv[a] : 0.0f;  A2[a].y = lo ? b2 : 0.0f;
  }
  __syncthreads();   // sCnt initialized before any push

  // ---- Phase A: slab test, 6 WMMAs per 16x16 tile, compact hits to LDS ----
  const int n0w = wave * (NVOX / 8);              // 512 voxels per wave
  for (int t = 0; t < (NVOX / 8) / 16; ++t) {     // 32 tiles
    const int n0 = n0w + t * 16;
    const int v  = n0 + row;
    float3 cv = reinterpret_cast<const float3*>(centers)[v];  // b96 load
    v2f B[3];
    B[0].x = lo ? cv.x : 0.0f;  B[0].y = lo ? 1.0f : 0.0f;
    B[1].x = lo ? cv.y : 0.0f;  B[1].y = lo ? 1.0f : 0.0f;
    B[2].x = lo ? cv.z : 0.0f;  B[2].y = lo ? 1.0f : 0.0f;

    v8f t1x = wmma4(A1[0], B[0]);
    v8f t2x = wmma4(A2[0], B[0]);
    v8f t1y = wmma4(A1[1], B[1]);
    v8f t2y = wmma4(A2[1], B[1]);
    v8f t1z = wmma4(A1[2], B[2]);
    v8f t2z = wmma4(A2[2], B[2]);

#pragma unroll
    for (int i = 0; i < 8; ++i) {
      float tn = fmaxf(fmaxf(fminf(t1x[i], t2x[i]), fminf(t1y[i], t2y[i])),
                       fminf(t1z[i], t2z[i]));
      float tf = fminf(fminf(fmaxf(t1x[i], t2x[i]), fmaxf(t1y[i], t2y[i])),
                       fmaxf(t1z[i], t2z[i]));
      const bool hit = (tn <= tf) && (tf > 0.0f);
      if (hit && tn < MISS_F) {                    // only real hits can place
        const int m = (lo ? 0 : 8) + i;            // ray row in block
        int slot = atomicAdd(&sCnt[m], 1);         // ds_add_rtn_u32
        if (slot < CAP) sKey[m][slot] = packKey(tn, n0 + row);
      }
    }
  }
  __syncthreads();

  // ---- Phase B: per-ray top-31 via wave32 u64-min butterfly ----
  // Wave w owns local rays 2w and 2w+1; <=64 candidates live in 2 regs/lane.
#pragma unroll
  for (int q = 0; q < 2; ++q) {
    const int rr = wave * 2 + q;
    int n = sCnt[rr];  n = n > CAP ? CAP : n;
    unsigned long long k0 = (lane      < n) ? sKey[rr][lane]      : ~0ull;
    unsigned long long k1 = (lane + 32 < n) ? sKey[rr][lane + 32] : ~0ull;
    for (int k = 0; k < KHITS; ++k) {
      unsigned long long m = k0 < k1 ? k0 : k1;
#pragma unroll
      for (int off = 16; off > 0; off >>= 1) {
        unsigned long long om =
            (unsigned long long)__shfl_xor((long long)m, off, 32);
        m = om < m ? om : m;
      }
      float d; int idx;
      if (m == ~0ull) {                            // padded miss slot
        d = MISS_F; idx = -1;
      } else {
        unsigned int u = (unsigned int)(m >> 32);
        unsigned int b = (u & 0x80000000u) ? (u ^ 0x80000000u) : ~u;
        d = __uint_as_float(b);
        idx = (int)(m & 0xFFFFFFFFu);
      }
      if (lane == 0) { sResD[rr][k] = d; sResI[rr][k] = idx; }
      if (k0 == m) k0 = ~0ull;                     // poison winner (idx unique)
      if (k1 == m) k1 = ~0ull;
    }
  }
  __syncthreads();

  // ---- Phase C: format outputs ----
  // d_out layout (flat, return order): out[R,31,9] | depth[R,31] | idx[R,31] | mask[R,31]
  const size_t RK = (size_t)numRays * KHITS;
  float* outMain  = out;
  float* outDepth = out + RK * 9;
  float* outIdx   = outDepth + RK;
  float* outMask  = outIdx + RK;

  for (int item = tid; item < RAYS_PER_WG * KHITS; item += 256) {
    const int rr   = item / KHITS;
    const int slot = item % KHITS;
    const int rg   = rBase + rr;

    float d = sResD[rr][slot];
    int   i = sResI[rr][slot];
    if (slot == KHITS - 1 || d >= MISS_F) i = -1;  // forced last slot + miss mask
    const bool msk = (i == -1);
    const int  ci  = msk ? 0 : i;

    float3 cc = reinterpret_cast<const float3*>(centers)[ci];
    const float* rp = rays + (size_t)rg * 6;
    const float ox = rp[0], oy = rp[1], oz = rp[2];
    const float dx = rp[3], dy = rp[4], dz = rp[5];

    const float px = msk ? fmaf(d, dx, ox) : (ox - cc.x);
    const float py = msk ? fmaf(d, dy, oy) : (oy - cc.y);
    const float pz = msk ? fmaf(d, dz, oz) : (oz - cc.z);

    const size_t base = (size_t)rg * KHITS + slot;
    float* po = outMain + base * 9;
    po[0] = px; po[1] = py; po[2] = pz;
    po[3] = dx; po[4] = dy; po[5] = dz;
    po[6] = cc.x; po[7] = cc.y; po[8] = cc.z;
    outDepth[base] = d;
    outIdx[base]   = (float)i;
    outMask[base]  = msk ? 1.0f : 0.0f;
  }
}

extern "C" void kernel_launch(void* const* d_in, const int* in_sizes, int n_in,
                              void* d_out, int out_size, void* d_ws, size_t ws_size,
                              hipStream_t stream) {
  (void)n_in; (void)out_size; (void)d_ws; (void)ws_size;
  const float* rays    = (const float*)d_in[0];
  const float* centers = (const float*)d_in[1];
  float* out = (float*)d_out;
  const int numRays = in_sizes[0] / 6;                 // 8192
  const int blocks  = (numRays + RAYS_PER_WG - 1) / RAYS_PER_WG;
  voxel_subdiv_kernel<<<blocks, 256, 0, stream>>>(rays, centers, out, numRays);
}